// GCN_19301583028828
// MI455X (gfx1250) — compile-verified
//
#include <hip/hip_runtime.h>
#include <hip/hip_bf16.h>
#include <math.h>

typedef float v2f __attribute__((ext_vector_type(2)));
typedef float v8f __attribute__((ext_vector_type(8)));

#define IN_C   128
#define HID_C  128
#define OUT_C  64
#define NBATCH 512

// ---------------------------------------------------------------------------
// Generic f32 WMMA GEMM:  C[M,N] = act(A[M,K] @ B[K,N] + bias)
// grid.x = M/16 row tiles; block = 32 * (N/16) threads, one wave per 16x16
// C tile. A tile (16 x K) staged in LDS with float4 loads; rows padded to
// 132 floats (16B-aligned, bank stride 4 -> conflict-free column reads).
// act: 0 = none, 1 = relu
// ---------------------------------------------------------------------------
__global__ __launch_bounds__(256) void gemm_f32_wmma(
    const float* __restrict__ A, const float* __restrict__ B,
    const float* __restrict__ bias, float* __restrict__ C,
    int M, int K, int N, int act)
{
    __shared__ float lds_a[16][132];           // K <= 128

    const int tid  = threadIdx.x;
    const int wave = tid >> 5;
    const int lane = tid & 31;
    const int row0 = blockIdx.x * 16;

    // cooperative float4 load of the 16 x K A tile
    const int nvec = (16 * K) >> 2;            // # of float4s
    const int kv   = K >> 2;                   // float4s per row
    for (int i = tid; i < nvec; i += blockDim.x) {
        int r  = i / kv;
        int k4 = (i - r * kv) << 2;
        float4 v = *(const float4*)(A + (size_t)(row0 + r) * K + k4);
        *(float4*)&lds_a[r][k4] = v;
    }
    __syncthreads();

    const int m    = lane & 15;          // A row in tile / B,N column in tile
    const int kh   = (lane >> 4) << 1;   // 0 for lanes 0-15, 2 for lanes 16-31
    const int col0 = wave * 16;

    v8f acc = {};
    for (int kb = 0; kb < K; kb += 4) {
        v2f a, b;
        a.x = lds_a[m][kb + kh];
        a.y = lds_a[m][kb + kh + 1];
        const float* Bp = B + (size_t)(kb + kh) * N + col0 + m;
        b.x = Bp[0];
        b.y = Bp[N];
        acc = __builtin_amdgcn_wmma_f32_16x16x4_f32(
            false, a, false, b, (short)0, acc, false, false);
    }

    const int   col   = col0 + m;
    const int   rbase = row0 + ((lane >> 4) << 3);   // +8 for lanes 16-31
    const float bv    = bias ? bias[col] : 0.0f;
#pragma unroll
    for (int v = 0; v < 8; ++v) {
        float x = acc[v] + bv;
        if (act == 1) x = fmaxf(x, 0.0f);
        C[(size_t)(rbase + v) * N + col] = x;
    }
}

// ---------------------------------------------------------------------------
// Small helpers
// ---------------------------------------------------------------------------
__global__ void fill_f32(float* __restrict__ p, float v, int n)
{
    int i = blockIdx.x * blockDim.x + threadIdx.x;
    if (i < n) p[i] = v;
}

__global__ void deg_edges(const int* __restrict__ dst, float* __restrict__ deg, int E)
{
    int e = blockIdx.x * blockDim.x + threadIdx.x;
    if (e < E) atomicAdd(&deg[dst[e]], 1.0f);
}

__global__ void rsqrt_k(const float* __restrict__ deg, float* __restrict__ dinv, int n)
{
    int i = blockIdx.x * blockDim.x + threadIdx.x;
    if (i < n) dinv[i] = rsqrtf(deg[i]);      // deg >= 1 (self-loops)
}

// h[i,c] = xw[i,c] * dinv[i]^2  (self-loop term; also fully initializes h)
// 32 threads per node, float4 per thread
__global__ __launch_bounds__(32) void self_loop_init(
    const float* __restrict__ xw, const float* __restrict__ dinv,
    float* __restrict__ h, int n)
{
    int node = blockIdx.x;
    int c4   = threadIdx.x << 2;
    float di = dinv[node];
    float s  = di * di;
    size_t idx = (size_t)node * HID_C + c4;
    float4 v = *(const float4*)(xw + idx);
    v.x *= s; v.y *= s; v.z *= s; v.w *= s;
    *(float4*)(h + idx) = v;
}

// h[dst] += xw[src] * dinv[src]*dinv[dst]
// 32 lanes per edge, float4 gather + 4 coalesced f32 atomics per lane
__global__ __launch_bounds__(256) void edge_scatter(
    const float* __restrict__ xw, const float* __restrict__ dinv,
    const int* __restrict__ src, const int* __restrict__ dst,
    float* __restrict__ h, int E)
{
    int e  = blockIdx.x * 8 + (threadIdx.x >> 5);
    int c4 = (threadIdx.x & 31) << 2;
    if (e < E) {
        int   s    = src[e];
        int   d    = dst[e];
        float norm = dinv[s] * dinv[d];
        float4 v = *(const float4*)(xw + (size_t)s * HID_C + c4);
        float* hp = h + (size_t)d * HID_C + c4;
        atomicAdd(hp + 0, v.x * norm);
        atomicAdd(hp + 1, v.y * norm);
        atomicAdd(hp + 2, v.z * norm);
        atomicAdd(hp + 3, v.w * norm);
    }
}

// h = gelu(h + b)  (exact erf form, matches jax.nn.gelu approximate=False)
__global__ __launch_bounds__(32) void bias_gelu(
    float* __restrict__ h, const float* __restrict__ b, int n)
{
    int node = blockIdx.x;
    int c4   = threadIdx.x << 2;
    size_t idx = (size_t)node * HID_C + c4;
    float4 v  = *(const float4*)(h + idx);
    float4 bb = *(const float4*)(b + c4);
    v.x += bb.x; v.y += bb.y; v.z += bb.z; v.w += bb.w;
    v.x = 0.5f * v.x * (1.0f + erff(v.x * 0.70710678118654752f));
    v.y = 0.5f * v.y * (1.0f + erff(v.y * 0.70710678118654752f));
    v.z = 0.5f * v.z * (1.0f + erff(v.z * 0.70710678118654752f));
    v.w = 0.5f * v.w * (1.0f + erff(v.w * 0.70710678118654752f));
    *(float4*)(h + idx) = v;
}

__global__ __launch_bounds__(32) void pool_accum(
    const float* __restrict__ h, const int* __restrict__ batch,
    float* __restrict__ sums, float* __restrict__ cnt, int n)
{
    int node = blockIdx.x;
    int c4   = threadIdx.x << 2;
    int b    = batch[node];
    float4 v = *(const float4*)(h + (size_t)node * HID_C + c4);
    float* sp = sums + (size_t)b * HID_C + c4;
    atomicAdd(sp + 0, v.x);
    atomicAdd(sp + 1, v.y);
    atomicAdd(sp + 2, v.z);
    atomicAdd(sp + 3, v.w);
    if (threadIdx.x == 0) atomicAdd(&cnt[b], 1.0f);
}

__global__ void pool_final(float* __restrict__ sums, const float* __restrict__ cnt, int n)
{
    int i = blockIdx.x * blockDim.x + threadIdx.x;
    if (i < n) sums[i] /= fmaxf(cnt[i / HID_C], 1.0f);
}

// ---------------------------------------------------------------------------
// Launch
// ---------------------------------------------------------------------------
extern "C" void kernel_launch(void* const* d_in, const int* in_sizes, int n_in,
                              void* d_out, int out_size, void* d_ws, size_t ws_size,
                              hipStream_t stream)
{
    const float* x     = (const float*)d_in[0];
    const int*   ei    = (const int*)d_in[1];
    const int*   batch = (const int*)d_in[2];
    const float* W[3]  = { (const float*)d_in[3], (const float*)d_in[5], (const float*)d_in[7] };
    const float* bv[3] = { (const float*)d_in[4], (const float*)d_in[6], (const float*)d_in[8] };
    const float* Wm1   = (const float*)d_in[9];
    const float* bm1   = (const float*)d_in[10];
    const float* Wm2   = (const float*)d_in[11];
    const float* bm2   = (const float*)d_in[12];

    const int N = in_sizes[0] / IN_C;     // 50000 (multiple of 16)
    const int E = in_sizes[1] / 2;        // 1200000
    const int* src = ei;                  // edge_index[0]
    const int* dst = ei + E;              // edge_index[1]

    // workspace carve-up
    char*  ws  = (char*)d_ws;
    size_t off = 0;
    auto carve = [&](size_t bytes) -> float* {
        float* p = (float*)(ws + off);
        off = (off + bytes + 255) & ~(size_t)255;
        return p;
    };
    float* deg  = carve((size_t)N * 4);
    float* dinv = carve((size_t)N * 4);
    float* buf0 = carve((size_t)N * HID_C * 4);   // xw
    float* buf1 = carve((size_t)N * HID_C * 4);   // h (doubles as next-layer x)
    float* psum = carve((size_t)NBATCH * HID_C * 4);
    float* pcnt = carve((size_t)NBATCH * 4);
    float* hid  = carve((size_t)NBATCH * HID_C * 4);

    // --- symmetric gcn normalization ---
    fill_f32<<<(N + 255) / 256, 256, 0, stream>>>(deg, 1.0f, N);   // self-loops
    deg_edges<<<(E + 255) / 256, 256, 0, stream>>>(dst, deg, E);
    rsqrt_k<<<(N + 255) / 256, 256, 0, stream>>>(deg, dinv, N);

    // --- 3 GCN layers ---
    const float* cur = x;
    for (int l = 0; l < 3; ++l) {
        gemm_f32_wmma<<<N / 16, 256, 0, stream>>>(cur, W[l], nullptr, buf0,
                                                  N, HID_C, HID_C, 0);
        self_loop_init<<<N, 32, 0, stream>>>(buf0, dinv, buf1, N);
        edge_scatter<<<(E + 7) / 8, 256, 0, stream>>>(buf0, dinv, src, dst, buf1, E);
        bias_gelu<<<N, 32, 0, stream>>>(buf1, bv[l], N);
        cur = buf1;
    }

    // --- global mean pool ---
    fill_f32<<<(NBATCH * HID_C + 255) / 256, 256, 0, stream>>>(psum, 0.0f, NBATCH * HID_C);
    fill_f32<<<(NBATCH + 255) / 256, 256, 0, stream>>>(pcnt, 0.0f, NBATCH);
    pool_accum<<<N, 32, 0, stream>>>(cur, batch, psum, pcnt, N);
    pool_final<<<(NBATCH * HID_C + 255) / 256, 256, 0, stream>>>(psum, pcnt, NBATCH * HID_C);

    // --- MLP head ---
    gemm_f32_wmma<<<NBATCH / 16, 256, 0, stream>>>(psum, Wm1, bm1, hid,
                                                   NBATCH, HID_C, HID_C, 1);
    gemm_f32_wmma<<<NBATCH / 16, 128, 0, stream>>>(hid, Wm2, bm2, (float*)d_out,
                                                   NBATCH, HID_C, OUT_C, 0);
}